// NghSampler_43954695307638
// MI455X (gfx1250) — compile-verified
//
#include <hip/hip_runtime.h>
#include <hip/hip_bf16.h>

typedef __attribute__((ext_vector_type(2))) float v2f;
typedef __attribute__((ext_vector_type(4))) float v4f;
typedef __attribute__((ext_vector_type(8))) float v8f;

#define NQ   7200      // number of query points (2 * 60 * 60)
#define KD   128       // feature channels
#define Wimg 512
#define Himg 512
#define GRID1 60       // grid points per dim per batch
#define PERB  3600     // GRID1*GRID1

// Workspace layout (float/int32 slots)
#define F1_OFF 0
#define D3_OFF (NQ * KD)
#define QX_OFF (2 * NQ * KD)
#define QY_OFF (QX_OFF + NQ)
#define DX_OFF (QY_OFF + NQ)
#define DY_OFF (DX_OFF + NQ)
#define X2_OFF (DY_OFF + NQ)
#define Y2_OFF (X2_OFF + NQ)

// ---------------------------------------------------------------------------
// Kernel 1: gather F1 / D3 rows into contiguous [n][128] panels, compute
// per-query xy2 / mask / qconf and batch-offset grid coordinates.
// One block (128 threads) per query point.
// ---------------------------------------------------------------------------
__global__ void __launch_bounds__(KD) prep_kernel(
    const float* __restrict__ feat1, const float* __restrict__ feat2,
    const float* __restrict__ conf1, const float* __restrict__ aflow,
    float* __restrict__ ws, float* __restrict__ mask_out,
    float* __restrict__ qconf_out) {
  int n = blockIdx.x;
  int k = threadIdx.x;
  int b  = n / PERB;
  int r  = n % PERB;
  int iy = r / GRID1;
  int ix = r % GRID1;
  int y = 16 + 8 * iy;
  int x = 16 + 8 * ix;

  size_t pix = (size_t)y * Wimg + x;
  size_t chan_stride = (size_t)Himg * Wimg;
  size_t base = ((size_t)b * KD) * chan_stride + pix;

  // gather f1 row (query features) and d3 row (db features, same grid)
  ws[F1_OFF + (size_t)n * KD + k] = feat1[base + (size_t)k * chan_stride];
  ws[D3_OFF + (size_t)n * KD + k] = feat2[base + (size_t)k * chan_stride];

  if (k == 0) {
    // conf1 shape (B,1,H,W)
    qconf_out[n] = conf1[(size_t)b * chan_stride + pix];
    // aflow shape (B,2,H,W)
    float ax = aflow[((size_t)b * 2 + 0) * chan_stride + pix];
    float ay = aflow[((size_t)b * 2 + 1) * chan_stride + pix];
    int x2 = (int)(ax + 0.5f);   // trunc toward zero, matches astype(int32)
    int y2 = (int)(ay + 0.5f);
    bool m = (x2 >= 0) && (y2 >= 0) && (x2 < Wimg) && (y2 < Himg);
    mask_out[n] = m ? 1.0f : 0.0f;
    ((int*)ws)[X2_OFF + n] = x2;
    ((int*)ws)[Y2_OFF + n] = y2;
    // batch-strided coordinates for the distance mask
    ws[QX_OFF + n] = (float)x2 + 512.0f * (float)b;
    ws[QY_OFF + n] = (float)y2 + 512.0f * (float)b;
    ws[DX_OFF + n] = (float)x  + 512.0f * (float)b;
    ws[DY_OFF + n] = (float)y  + 512.0f * (float)b;
  }
}

// ---------------------------------------------------------------------------
// Kernel 2: gt fill (1.0 for first P columns of each row, else 0.0)
// ---------------------------------------------------------------------------
__global__ void gt_fill_kernel(float* __restrict__ gt, long long total,
                               int cols, int P) {
  long long i = (long long)blockIdx.x * blockDim.x + threadIdx.x;
  long long stride = (long long)gridDim.x * blockDim.x;
  for (; i < total; i += stride) {
    int c = (int)(i % cols);
    gt[i] = (c < P) ? 1.0f : 0.0f;
  }
}

// ---------------------------------------------------------------------------
// Kernel 3: pos/neg scores. One block per query; f1 row staged in LDS;
// threads 0..P+NN-1 each compute one offset's 128-dot against feat2.
// ---------------------------------------------------------------------------
__global__ void __launch_bounds__(128) posneg_kernel(
    const float* __restrict__ ws, const float* __restrict__ feat2,
    const int* __restrict__ pos_off, const int* __restrict__ neg_off,
    float* __restrict__ out, int cols, int P, int NN) {
  __shared__ float f1[KD];
  int n = blockIdx.x;
  int t = threadIdx.x;
  f1[t] = ws[F1_OFF + (size_t)n * KD + t];
  __syncthreads();
  int npair = P + NN;
  if (t >= npair) return;

  int b  = n / PERB;
  int x2 = ((const int*)ws)[X2_OFF + n];
  int y2 = ((const int*)ws)[Y2_OFF + n];
  int ox, oy;
  if (t < P) { ox = pos_off[t];      oy = pos_off[P + t]; }
  else       { ox = neg_off[t - P];  oy = neg_off[NN + (t - P)]; }
  int xc = min(max(x2 + ox, 0), Wimg - 1);
  int yc = min(max(y2 + oy, 0), Himg - 1);

  size_t chan_stride = (size_t)Himg * Wimg;
  const float* f2p = feat2 + ((size_t)b * KD) * chan_stride +
                     (size_t)yc * Wimg + xc;
  float s = 0.0f;
#pragma unroll 8
  for (int k = 0; k < KD; ++k) s += f1[k] * f2p[(size_t)k * chan_stride];
  out[(size_t)n * cols + t] = s;
}

// ---------------------------------------------------------------------------
// Kernel 4: dscores GEMM  C[7200x7200] = F1(7200x128) x D3^T  via
// V_WMMA_F32_16X16X4_F32, with the grid-distance mask fused in the epilogue.
// Block: 192 threads = 6 wave32s in a 3(M) x 2(N) wave grid; tile 96x96;
// K=128 fully resident in LDS (padded stride 132 floats -> conflict-free).
// Panels are filled with GLOBAL_LOAD_ASYNC_TO_LDS_B128 (ASYNCcnt path),
// avoiding the VGPR round-trip entirely.
// ---------------------------------------------------------------------------
#define TILE   96
#define LDSTR  132   // 128 + 4 pad; row-to-row bank step = 4

__global__ void __launch_bounds__(192) gemm_wmma_kernel(
    const float* __restrict__ ws, float* __restrict__ out, int cols,
    int col0) {
  extern __shared__ float smem[];
  float* As = smem;                  // [96][132]
  float* Bs = smem + TILE * LDSTR;   // [96][132]

  const float* F1 = ws + F1_OFF;
  const float* D3 = ws + D3_OFF;
  const float* QX = ws + QX_OFF;
  const float* QY = ws + QY_OFF;
  const float* DX = ws + DX_OFF;
  const float* DY = ws + DY_OFF;

  int m0 = blockIdx.y * TILE;   // query rows
  int n0 = blockIdx.x * TILE;   // db columns
  int t  = threadIdx.x;

  // Cooperative async fill: each panel is 96 rows x 32 float4s = 3072 b128s.
  // GVS addressing: SGPR-pair base (kernel-arg uniform) + per-lane byte offset.
  // VDST = LDS byte address (low 32 bits of the flat shared-pointer address).
  for (int i = t; i < TILE * (KD / 4); i += 192) {
    int row = i >> 5;
    int c4  = (i & 31) << 2;
    unsigned goff = (unsigned)((m0 + row) * KD + c4) * 4u;   // bytes into F1
    unsigned hoff = (unsigned)((n0 + row) * KD + c4) * 4u;   // bytes into D3
    unsigned la = (unsigned)(uintptr_t)(As + row * LDSTR + c4);
    unsigned lb = (unsigned)(uintptr_t)(Bs + row * LDSTR + c4);
    asm volatile("global_load_async_to_lds_b128 %0, %1, %2"
                 :
                 : "v"(la), "v"(goff), "s"(F1)
                 : "memory");
    asm volatile("global_load_async_to_lds_b128 %0, %1, %2"
                 :
                 : "v"(lb), "v"(hoff), "s"(D3)
                 : "memory");
  }
  asm volatile("s_wait_asynccnt 0x0" ::: "memory");
  __syncthreads();

  int w    = t >> 5;        // wave id 0..5
  int lane = t & 31;
  int wm   = w % 3;         // M sub-tile (32 rows each)
  int wn   = w / 3;         // N sub-tile (48 cols each)
  int lo   = lane & 15;
  int hi   = lane >> 4;

  v8f z = {0.f, 0.f, 0.f, 0.f, 0.f, 0.f, 0.f, 0.f};
  v8f acc[2][3];
#pragma unroll
  for (int i = 0; i < 2; ++i)
#pragma unroll
    for (int j = 0; j < 3; ++j) acc[i][j] = z;

  // Fragment bases. A(16x4): lane lo holds row m, VGPR pair = K {2*hi, 2*hi+1}.
  // B(4x16): lane lo holds col n, VGPR pair = K rows {2*hi, 2*hi+1} == D3[n][k..].
  const float* a0p = As + (32 * wm + lo) * LDSTR + 2 * hi;
  const float* a1p = As + (32 * wm + 16 + lo) * LDSTR + 2 * hi;
  const float* b0p = Bs + (48 * wn + lo) * LDSTR + 2 * hi;
  const float* b1p = Bs + (48 * wn + 16 + lo) * LDSTR + 2 * hi;
  const float* b2p = Bs + (48 * wn + 32 + lo) * LDSTR + 2 * hi;

#pragma unroll 4
  for (int k = 0; k < KD; k += 4) {
    v2f a0 = *(const v2f*)(a0p + k);
    v2f a1 = *(const v2f*)(a1p + k);
    v2f b0 = *(const v2f*)(b0p + k);
    v2f b1 = *(const v2f*)(b1p + k);
    v2f b2 = *(const v2f*)(b2p + k);
    acc[0][0] = __builtin_amdgcn_wmma_f32_16x16x4_f32(false, a0, false, b0,
                                                      (short)0, acc[0][0],
                                                      false, false);
    acc[0][1] = __builtin_amdgcn_wmma_f32_16x16x4_f32(false, a0, false, b1,
                                                      (short)0, acc[0][1],
                                                      false, false);
    acc[0][2] = __builtin_amdgcn_wmma_f32_16x16x4_f32(false, a0, false, b2,
                                                      (short)0, acc[0][2],
                                                      false, false);
    acc[1][0] = __builtin_amdgcn_wmma_f32_16x16x4_f32(false, a1, false, b0,
                                                      (short)0, acc[1][0],
                                                      false, false);
    acc[1][1] = __builtin_amdgcn_wmma_f32_16x16x4_f32(false, a1, false, b1,
                                                      (short)0, acc[1][1],
                                                      false, false);
    acc[1][2] = __builtin_amdgcn_wmma_f32_16x16x4_f32(false, a1, false, b2,
                                                      (short)0, acc[1][2],
                                                      false, false);
  }

  // Epilogue: C/D layout -> VGPR v, lanes 0-15 : M = v, lanes 16-31 : M = v+8.
  // Fuse dscores = where(dis < 4, 0, dscores)  via squared distance.
#pragma unroll
  for (int im = 0; im < 2; ++im) {
#pragma unroll
    for (int jn = 0; jn < 3; ++jn) {
      int ncol = n0 + 48 * wn + 16 * jn + lo;
      float cdx = DX[ncol];
      float cdy = DY[ncol];
#pragma unroll
      for (int v = 0; v < 8; ++v) {
        int mrow = m0 + 32 * wm + 16 * im + 8 * hi + v;
        float fx = QX[mrow] - cdx;
        float fy = QY[mrow] - cdy;
        float val = acc[im][jn][v];
        if (fx * fx + fy * fy < 16.0f) val = 0.0f;
        out[(size_t)mrow * cols + col0 + ncol] = val;
      }
    }
  }
}

// ---------------------------------------------------------------------------
extern "C" void kernel_launch(void* const* d_in, const int* in_sizes, int n_in,
                              void* d_out, int out_size, void* d_ws,
                              size_t ws_size, hipStream_t stream) {
  const float* feat1 = (const float*)d_in[0];
  const float* feat2 = (const float*)d_in[1];
  const float* conf1 = (const float*)d_in[2];
  // d_in[3] = conf2 (unused by reference)
  const float* aflow = (const float*)d_in[4];
  const int* pos_off = (const int*)d_in[5];
  const int* neg_off = (const int*)d_in[6];

  int P  = in_sizes[5] / 2;   // 13
  int NN = in_sizes[6] / 2;   // 104
  int cols = P + NN + NQ;     // 7317

  float* ws  = (float*)d_ws;
  float* out = (float*)d_out;                    // scores
  long long scoreN = (long long)NQ * cols;
  float* gt        = out + scoreN;               // gt (as 0/1 floats)
  float* mask_out  = out + 2 * scoreN;           // mask (as 0/1 floats)
  float* qconf_out = mask_out + NQ;              // qconf

  prep_kernel<<<NQ, KD, 0, stream>>>(feat1, feat2, conf1, aflow, ws, mask_out,
                                     qconf_out);

  gt_fill_kernel<<<4096, 256, 0, stream>>>(gt, scoreN, cols, P);

  posneg_kernel<<<NQ, 128, 0, stream>>>(ws, feat2, pos_off, neg_off, out, cols,
                                        P, NN);

  size_t shmem = (size_t)2 * TILE * LDSTR * sizeof(float);  // ~101 KB
  hipFuncSetAttribute((const void*)gemm_wmma_kernel,
                      hipFuncAttributeMaxDynamicSharedMemorySize, (int)shmem);
  dim3 grid(NQ / TILE, NQ / TILE);   // 75 x 75
  gemm_wmma_kernel<<<grid, 192, shmem, stream>>>(ws, out, cols, P + NN);
}